// AttentionEmbedding_22789096472775
// MI455X (gfx1250) — compile-verified
//
#include <hip/hip_runtime.h>
#include <stdint.h>

// CDNA5 / gfx1250: wave32, WMMA f32 16x16x4 (A,B = v2f per lane; C/D = v8f).
typedef __attribute__((ext_vector_type(2))) float v2f;
typedef __attribute__((ext_vector_type(8))) float v8f;

#define BQ     16                  // rows/cols per WMMA tile
#define CDIM   256                 // channels (GEMM K dimension)
#define HWDIM  4096                // keys/queries per batch
#define VIS    1024                // visible corner of S
#define MROWS  128                 // query rows per workgroup (8 M-tiles)
#define NMT    (MROWS / BQ)        // 8, one M-tile per wave
#define NPASS  (HWDIM / 32)        // 128 passes, 2 key tiles per pass
#define TILE_F (CDIM * BQ)         // 4096 floats per staged tile
#define QS_F   (NMT * TILE_F)      // 32768 floats = 128 KB
#define KS_F   (2 * TILE_F)        // 8192 floats  = 32 KB

// Monotone map: f32 -> u32 preserving total order.
__device__ __forceinline__ unsigned int mono_f32(float f) {
    unsigned int u = __float_as_uint(f);
    return (u & 0x80000000u) ? ~u : (u | 0x80000000u);
}

// Stage one c-pair row (c0=2j, c1=2j+1) x 16 columns into c-paired LDS layout:
// dst[m*2+0] = row_c0[m], dst[m*2+1] = row_c1[m]  (b128 global in, b128 LDS out).
__device__ __forceinline__ void stage_pair_row(
    const float* __restrict__ g0, const float* __restrict__ g1,
    float* __restrict__ dst)
{
    float r0[16], r1[16];
    #pragma unroll
    for (int e = 0; e < 4; ++e) {
        float4 x0 = ((const float4*)g0)[e];
        float4 x1 = ((const float4*)g1)[e];
        r0[4*e+0]=x0.x; r0[4*e+1]=x0.y; r0[4*e+2]=x0.z; r0[4*e+3]=x0.w;
        r1[4*e+0]=x1.x; r1[4*e+1]=x1.y; r1[4*e+2]=x1.z; r1[4*e+3]=x1.w;
    }
    #pragma unroll
    for (int d = 0; d < 8; ++d)
        ((float4*)dst)[d] = make_float4(r0[2*d], r1[2*d], r0[2*d+1], r1[2*d+1]);
}

// One pass: stage 2 key tiles into LDS, then every wave runs its M-tile
// against both (two independent WMMA chains sharing each A operand).
template<bool STORE>
__device__ __forceinline__ void pass(
    float* __restrict__ Qs, float* __restrict__ Ks,
    const float* __restrict__ Kb,
    int kp, int tid, int wave, int m, int ch,
    float* __restrict__ svb,           // Svis + b*VIS^2 + (q0+wave*16)*VIS + m
    float bval[8], int bidx[8])
{
    __syncthreads();                   // previous compute done before overwrite
    {   // 256 pair-rows (2 tiles x 128 c-pairs), one per thread.
        const int tile = tid >> 7, j = tid & 127;
        const int kcol = kp * 32 + tile * BQ;
        const float* g0 = Kb + (size_t)(2*j)     * HWDIM + kcol;
        const float* g1 = Kb + (size_t)(2*j + 1) * HWDIM + kcol;
        stage_pair_row(g0, g1, Ks + tile * TILE_F + j * 32);
        if (kp + 1 < NPASS) __builtin_prefetch(g0 + 32, 0, 1);
    }
    __syncthreads();

    v8f acc0 = {}, acc1 = {};
    // Independent pointer inductions per LDS stream so each keeps its own
    // DS base register with small immediate offsets (=> ds_load_2addr_b64),
    // instead of being rebased onto one pointer with unfoldable deltas.
    const int lofs = ch * 32 + m * 2;
    const float* ap  = Qs + wave * TILE_F + lofs;
    const float* k0p = Ks + lofs;
    const float* k1p = k0p + TILE_F;
    #pragma unroll 8
    for (int i = 0; i < CDIM / 4; ++i) {
        v2f a  = *(const v2f*)ap;
        v2f b0 = *(const v2f*)k0p;
        v2f b1 = *(const v2f*)k1p;
        acc0 = __builtin_amdgcn_wmma_f32_16x16x4_f32(
            false, a, false, b0, (short)0, acc0, false, false);
        acc1 = __builtin_amdgcn_wmma_f32_16x16x4_f32(
            false, a, false, b1, (short)0, acc1, false, false);
        ap += 64; k0p += 64; k1p += 64;   // next 4 channels (paired layout)
    }

    // Lane sees indices in strictly increasing order across passes/chains,
    // so strict '>' keeps the first occurrence of the max (jnp.argmax).
    const int k0 = kp * 32 + m;
    #pragma unroll
    for (int r = 0; r < 8; ++r) {
        const float v0 = acc0[r], v1 = acc1[r];
        const bool g0 = v0 > bval[r];
        bval[r] = g0 ? v0 : bval[r];
        bidx[r] = g0 ? k0 : bidx[r];
        const bool g1 = v1 > bval[r];
        bval[r] = g1 ? v1 : bval[r];
        bidx[r] = g1 ? (k0 + BQ) : bidx[r];
        if (STORE) {
            const int row = r + 8 * ch;   // D layout: VGPR r -> row r / r+8
            svb[(size_t)row * VIS + kp * 32]      = v0;
            svb[(size_t)row * VIS + kp * 32 + BQ] = v1;
        }
    }
}

__global__ __launch_bounds__(256)
void attn_s_argmax_kernel(const float* __restrict__ Qg,
                          const float* __restrict__ Kg,
                          float* __restrict__ Svis,
                          long long* __restrict__ Hidx)
{
    __shared__ float lds[QS_F + KS_F];   // 160 KB -> 2 workgroups per WGP
    float* Qs = lds;
    float* Ks = lds + QS_F;

    const int b   = blockIdx.y;
    const int bx  = blockIdx.x;          // 0..31 query strips
    const int q0  = bx * MROWS;
    const int tid = threadIdx.x;
    const int wave = tid >> 5, lane = tid & 31;
    const int m = lane & 15, ch = lane >> 4;

    // ---- Stage the 128x256 Q strip once (1024 pair-rows, 4 per thread).
    {
        const float* Qb = Qg + (size_t)b * CDIM * HWDIM;
        #pragma unroll
        for (int i = 0; i < 4; ++i) {
            const int r  = tid + 256 * i;
            const int mt = r >> 7, j = r & 127;
            const int qcol = q0 + mt * BQ;
            stage_pair_row(Qb + (size_t)(2*j)     * HWDIM + qcol,
                           Qb + (size_t)(2*j + 1) * HWDIM + qcol,
                           Qs + mt * TILE_F + j * 32);
        }
    }   // first pass() opens with __syncthreads(), covering Q staging too

    float bval[8];
    int   bidx[8];
    #pragma unroll
    for (int r = 0; r < 8; ++r) { bval[r] = -__builtin_inff(); bidx[r] = 0; }

    const float* Kb = Kg + (size_t)b * CDIM * HWDIM;
    float* svb = Svis + (size_t)b * VIS * VIS
                      + (size_t)(q0 + wave * BQ) * VIS + m;

    // Block-uniform visibility split: strips 0..7 store passes 0..31.
    if (bx < VIS / MROWS) {
        for (int kp = 0; kp < VIS / 32; ++kp)
            pass<true >(Qs, Ks, Kb, kp, tid, wave, m, ch, svb, bval, bidx);
        for (int kp = VIS / 32; kp < NPASS; ++kp)
            pass<false>(Qs, Ks, Kb, kp, tid, wave, m, ch, svb, bval, bidx);
    } else {
        for (int kp = 0; kp < NPASS; ++kp)
            pass<false>(Qs, Ks, Kb, kp, tid, wave, m, ch, svb, bval, bidx);
    }

    // ---- Merge per-lane maxima across the 16 lanes holding each row.
    __syncthreads();
    unsigned long long* Lmax = (unsigned long long*)Ks;  // K buffer is dead
    if (tid < MROWS) Lmax[tid] = 0ull;
    __syncthreads();
    #pragma unroll
    for (int r = 0; r < 8; ++r) {
        const int row = wave * BQ + r + 8 * ch;
        const unsigned long long key =
            ((unsigned long long)mono_f32(bval[r]) << 32) |
            (unsigned long long)(HWDIM - 1 - bidx[r]);   // first-max tie-break
        atomicMax(&Lmax[row], key);
    }
    __syncthreads();
    if (tid < MROWS) {
        const unsigned int inv = (unsigned int)(Lmax[tid] & 0xFFFFFFFFull);
        Hidx[(size_t)b * HWDIM + q0 + tid] = (long long)(HWDIM - 1 - inv);
    }
}

extern "C" void kernel_launch(void* const* d_in, const int* in_sizes, int n_in,
                              void* d_out, int out_size, void* d_ws, size_t ws_size,
                              hipStream_t stream) {
    (void)in_sizes; (void)n_in; (void)d_ws; (void)ws_size; (void)out_size;
    const float* Q = (const float*)d_in[0];
    const float* K = (const float*)d_in[1];
    // d_in[2] (V) is unused by the reference outputs.
    float* Svis = (float*)d_out;                                           // 4*1024*1024 f32
    long long* Hidx = (long long*)((float*)d_out + (size_t)4 * VIS * VIS); // 4*4096 i64

    dim3 grid(HWDIM / MROWS, 4);   // 32 query strips x 4 batches
    dim3 block(256);               // 8 wave32
    attn_s_argmax_kernel<<<grid, block, 0, stream>>>(Q, K, Svis, Hidx);
}